// GCNLayer_77661598646422
// MI455X (gfx1250) — compile-verified
//
#include <hip/hip_runtime.h>

typedef __attribute__((ext_vector_type(2))) float v2f;
typedef __attribute__((ext_vector_type(8))) float v8f;

#define IN_DIM  128
#define OUT_DIM 64
#define NHEADS  4

// ---------------------------------------------------------------------------
// Kernel 1: Weff[d][f] = (1/4) * sum_k W[k][d][f]   (8192 elements)
// ---------------------------------------------------------------------------
__global__ void gcn_reduce_w(const float* __restrict__ W, float* __restrict__ Weff) {
    int i = blockIdx.x * blockDim.x + threadIdx.x;
    if (i < IN_DIM * OUT_DIM) {
        float s = 0.f;
#pragma unroll
        for (int k = 0; k < NHEADS; ++k) s += W[k * IN_DIM * OUT_DIM + i];
        Weff[i] = s * 0.25f;
    }
}

// ---------------------------------------------------------------------------
// Kernel 2: hp[N,64] = h[N,128] @ Weff[128,64] via V_WMMA_F32_16X16X4_F32.
// 256 threads = 8 waves; each wave owns one 16-row tile x all 4 col tiles.
// Weff staged in LDS in K-pair-interleaved layout so every B fragment is a
// single aligned ds_load_b64 landing directly in an even VGPR pair:
//   lb2[p*128 + n*2 + 0] = Weff[2p  ][n]
//   lb2[p*128 + n*2 + 1] = Weff[2p+1][n]        (p = k/2, n = 0..63)
// ---------------------------------------------------------------------------
__global__ void __launch_bounds__(256)
gcn_gemm_wmma(const float* __restrict__ h, const float* __restrict__ Weff,
              float* __restrict__ hp, int nRowTiles, int nNodes) {
    __shared__ float lb2[IN_DIM * OUT_DIM];  // 32 KB, pair-interleaved
    for (int i = threadIdx.x; i < IN_DIM * OUT_DIM; i += 256) {
        int k = i >> 6;          // 0..127  (coalesced global read)
        int n = i & 63;          // 0..63
        lb2[(k >> 1) * 128 + n * 2 + (k & 1)] = Weff[i];
    }
    __syncthreads();

    const int wave = threadIdx.x >> 5;
    const int lane = threadIdx.x & 31;
    const int rt   = blockIdx.x * 8 + wave;
    if (rt >= nRowTiles) return;           // whole-wave uniform exit (EXEC stays all-1s)

    const int half = lane >> 4;            // 0: K pair {0,1}, 1: K pair {2,3}
    const int m    = lane & 15;            // row-in-tile for A, col-in-tile for B/C/D
    int row = rt * 16 + m;
    if (row >= nNodes) row = nNodes - 1;   // clamp loads (stores guarded below)
    const float* hrow = h + (size_t)row * IN_DIM;

    v8f acc0 = {}, acc1 = {}, acc2 = {}, acc3 = {};

#pragma unroll 4
    for (int kk = 0; kk < IN_DIM / 4; ++kk) {
        const int k0 = kk * 4 + half * 2;
        // A fragment: 16x4 f32 -> 2 VGPRs per lane (one global b64 load)
        float2 av = *(const float2*)(hrow + k0);
        v2f a; a.x = av.x; a.y = av.y;
        // B fragments: one aligned ds_load_b64 each (pair-interleaved layout)
        const int p = kk * 2 + half;       // K-pair index
        const float* bbase = lb2 + p * 128 + m * 2;
        v2f b0 = *(const v2f*)(bbase +  0);
        v2f b1 = *(const v2f*)(bbase + 32);
        v2f b2 = *(const v2f*)(bbase + 64);
        v2f b3 = *(const v2f*)(bbase + 96);

        acc0 = __builtin_amdgcn_wmma_f32_16x16x4_f32(false, a, false, b0, (short)0, acc0, false, false);
        acc1 = __builtin_amdgcn_wmma_f32_16x16x4_f32(false, a, false, b1, (short)0, acc1, false, false);
        acc2 = __builtin_amdgcn_wmma_f32_16x16x4_f32(false, a, false, b2, (short)0, acc2, false, false);
        acc3 = __builtin_amdgcn_wmma_f32_16x16x4_f32(false, a, false, b3, (short)0, acc3, false, false);
    }

    // C/D layout: lane group (0-15 / 16-31) -> M = v + 8*half, N = m
#pragma unroll
    for (int v = 0; v < 8; ++v) {
        const int orow = rt * 16 + v + 8 * half;
        if (orow < nNodes) {
            float* o = hp + (size_t)orow * OUT_DIM + m;
            o[0]  = acc0[v];
            o[16] = acc1[v];
            o[32] = acc2[v];
            o[48] = acc3[v];
        }
    }
}

// ---------------------------------------------------------------------------
// Kernel 3: zero the aggregation buffer (d_out), float4-wide
// ---------------------------------------------------------------------------
__global__ void gcn_zero_f4(float4* __restrict__ p, int n4) {
    int i = blockIdx.x * blockDim.x + threadIdx.x;
    if (i < n4) p[i] = make_float4(0.f, 0.f, 0.f, 0.f);
}

// ---------------------------------------------------------------------------
// Kernel 4: scatter-sum. 16 threads per edge, each handles 4 features.
// float4 gather from hp (L2-resident) + 4x global_atomic_add_f32.
// ---------------------------------------------------------------------------
__global__ void gcn_scatter(const float* __restrict__ hp, const int* __restrict__ src,
                            const int* __restrict__ dst, float* __restrict__ agg,
                            int nEdges) {
    int gid = blockIdx.x * blockDim.x + threadIdx.x;
    int e = gid >> 4;
    if (e >= nEdges) return;
    int q = gid & 15;
    int s = src[e];
    int d = dst[e];
    float4 v = ((const float4*)(hp + (size_t)s * OUT_DIM))[q];
    float* base = agg + (size_t)d * OUT_DIM + q * 4;
    atomicAdd(base + 0, v.x);
    atomicAdd(base + 1, v.y);
    atomicAdd(base + 2, v.z);
    atomicAdd(base + 3, v.w);
}

// ---------------------------------------------------------------------------
// Kernel 5: out = relu(agg + b), in place on d_out, float4-wide
// ---------------------------------------------------------------------------
__global__ void gcn_bias_relu(float* __restrict__ out, const float* __restrict__ b,
                              int nNodes) {
    int gid = blockIdx.x * blockDim.x + threadIdx.x;
    if (gid >= nNodes * (OUT_DIM / 4)) return;
    int q = gid & (OUT_DIM / 4 - 1);
    float4 v  = ((float4*)out)[gid];
    float4 bb = ((const float4*)b)[q];
    v.x = fmaxf(v.x + bb.x, 0.f);
    v.y = fmaxf(v.y + bb.y, 0.f);
    v.z = fmaxf(v.z + bb.z, 0.f);
    v.w = fmaxf(v.w + bb.w, 0.f);
    ((float4*)out)[gid] = v;
}

// ---------------------------------------------------------------------------
extern "C" void kernel_launch(void* const* d_in, const int* in_sizes, int n_in,
                              void* d_out, int out_size, void* d_ws, size_t ws_size,
                              hipStream_t stream) {
    const float* h   = (const float*)d_in[0];
    const float* W   = (const float*)d_in[1];
    const float* b   = (const float*)d_in[2];
    const int*   src = (const int*)d_in[3];
    const int*   dst = (const int*)d_in[4];
    float* out = (float*)d_out;

    const int nNodes = in_sizes[0] / IN_DIM;
    const int nEdges = in_sizes[3];

    float* Weff = (float*)d_ws;                 // 128*64 floats = 32 KB
    float* hp   = Weff + IN_DIM * OUT_DIM;      // nNodes*64 floats = 12.8 MB

    // 1) Weff = mean_k W_k
    gcn_reduce_w<<<(IN_DIM * OUT_DIM + 255) / 256, 256, 0, stream>>>(W, Weff);

    // 2) hp = h @ Weff   (WMMA)
    const int nRowTiles = (nNodes + 15) / 16;
    gcn_gemm_wmma<<<(nRowTiles + 7) / 8, 256, 0, stream>>>(h, Weff, hp, nRowTiles, nNodes);

    // 3) zero accumulator (d_out)
    const int n4 = nNodes * OUT_DIM / 4;
    gcn_zero_f4<<<(n4 + 255) / 256, 256, 0, stream>>>((float4*)out, n4);

    // 4) scatter-sum hp[src] into d_out[dst]
    const int sthreads = nEdges * 16;
    gcn_scatter<<<(sthreads + 255) / 256, 256, 0, stream>>>(hp, src, dst, out, nEdges);

    // 5) out = relu(out + b)
    const int rthreads = nNodes * (OUT_DIM / 4);
    gcn_bias_relu<<<(rthreads + 255) / 256, 256, 0, stream>>>(out, b, nNodes);
}